// RungeKuttaIntegratorCell_89163521065700
// MI455X (gfx1250) — compile-verified
//
#include <hip/hip_runtime.h>

// Problem constants (from the reference)
#define T_STEPS 4096
#define BATCH   4096
#define SEGS    128
#define SEG_LEN (T_STEPS / SEGS)   // 32 steps per segment
#define BLK     256
#define NSLOT   8                  // async pipeline depth (LDS ring slots)
#define STEP_BYTES ((unsigned long long)BATCH * 2u * 4u)   // u stride per step

static_assert((SEG_LEN & (SEG_LEN - 1)) == 0, "SEG_LEN must be a power of two");
static_assert(T_STEPS % SEGS == 0, "segments must tile T");
static_assert(BATCH % BLK == 0, "block must tile batch");
static_assert(SEG_LEN >= NSLOT, "pipeline deeper than segment");

typedef __attribute__((ext_vector_type(2))) float f32x2;
typedef __attribute__((ext_vector_type(4))) float f32x4;

// ---------------------------------------------------------------------------
// CDNA5 async global->LDS helpers (gfx1250 ASYNCcnt path).
// GVS form: scalar 64-bit base (SALU-maintained) + loop-invariant 32-bit
// per-lane offset, so the hot loop spends no VALU on address formation.
// ---------------------------------------------------------------------------
__device__ __forceinline__ void async_copy_b64_gvs(unsigned lds_off, unsigned voff,
                                                   unsigned long long sbase) {
    asm volatile("global_load_async_to_lds_b64 %0, %1, %2"
                 :
                 : "v"(lds_off), "v"(voff), "s"(sbase)
                 : "memory");
}
// Steady-state wait: allow NSLOT-1 outstanding (oldest slot is then staged).
__device__ __forceinline__ void wait_async_steady() {
    asm volatile("s_wait_asynccnt 0x7" ::: "memory");
}
// Drain waits (cold epilogue only).
__device__ __forceinline__ void wait_async_le(int k) {
    switch (k) {
    case 0: asm volatile("s_wait_asynccnt 0x0" ::: "memory"); break;
    case 1: asm volatile("s_wait_asynccnt 0x1" ::: "memory"); break;
    case 2: asm volatile("s_wait_asynccnt 0x2" ::: "memory"); break;
    case 3: asm volatile("s_wait_asynccnt 0x3" ::: "memory"); break;
    case 4: asm volatile("s_wait_asynccnt 0x4" ::: "memory"); break;
    case 5: asm volatile("s_wait_asynccnt 0x5" ::: "memory"); break;
    default: asm volatile("s_wait_asynccnt 0x6" ::: "memory"); break;
    }
}

// Streaming (non-temporal) stores: keep write streams from evicting the
// L2-resident u[] that pass 3 re-reads.
__device__ __forceinline__ void store_nt_f2(float2* p, float a, float b) {
    f32x2 v; v.x = a; v.y = b;
    __builtin_nontemporal_store(v, (f32x2*)p);
}
__device__ __forceinline__ void store_nt_f4(float4* p, float a, float b, float c, float d) {
    f32x4 v; v.x = a; v.y = b; v.z = c; v.w = d;
    __builtin_nontemporal_store(v, (f32x4*)p);
}

// ---------------------------------------------------------------------------
// Discrete update x_{t+1} = Ad x_t + Bd u_t implied by the reference's
// (simplified) RK4 stage arithmetic. All fp32, wave-uniform (lands on SALU FP).
// ---------------------------------------------------------------------------
__device__ __forceinline__ void compute_AdBd(const float* __restrict__ Ct,
                                             float A[4][4], float B[4][2]) {
    const float Mi0 = 1.0f / 20.0f, Mi1 = 1.0f / 10.0f;   // M_DIAG = [20,10]
    const float h = 0.002f, h2 = 0.5f * h * h;
    // K = _ck([2000,1000,5000]) = [[3000,-1000],[-1000,6000]]
    const float k00 = 3000.0f, k01 = -1000.0f, k10 = -1000.0f, k11 = 6000.0f;
    const float c0 = Ct[0], c1 = Ct[1], c2 = Ct[2];
    const float C00 = c0 + c1, C01 = -c1, C10 = -c1, C11 = c1 + c2;
    // P = -Minv K ; Q = -Minv C
    const float p00 = -Mi0 * k00, p01 = -Mi0 * k01, p10 = -Mi1 * k10, p11 = -Mi1 * k11;
    const float q00 = -Mi0 * C00, q01 = -Mi0 * C01, q10 = -Mi1 * C10, q11 = -Mi1 * C11;
    const float qp00 = q00 * p00 + q01 * p10, qp01 = q00 * p01 + q01 * p11;
    const float qp10 = q10 * p00 + q11 * p10, qp11 = q10 * p01 + q11 * p11;
    const float qq00 = q00 * q00 + q01 * q10, qq01 = q00 * q01 + q01 * q11;
    const float qq10 = q10 * q00 + q11 * q10, qq11 = q10 * q01 + q11 * q11;

    A[0][0] = 1.0f + h2 * p00;  A[0][1] = h2 * p01;         A[0][2] = h + h2 * q00;  A[0][3] = h2 * q01;
    A[1][0] = h2 * p10;         A[1][1] = 1.0f + h2 * p11;  A[1][2] = h2 * q10;      A[1][3] = h + h2 * q11;
    A[2][0] = h * p00 + h2 * qp00;  A[2][1] = h * p01 + h2 * qp01;
    A[2][2] = 1.0f + h * q00 + h2 * (p00 + qq00);  A[2][3] = h * q01 + h2 * (p01 + qq01);
    A[3][0] = h * p10 + h2 * qp10;  A[3][1] = h * p11 + h2 * qp11;
    A[3][2] = h * q10 + h2 * (p10 + qq10);  A[3][3] = 1.0f + h * q11 + h2 * (p11 + qq11);

    B[0][0] = h2 * Mi0;                 B[0][1] = 0.0f;
    B[1][0] = 0.0f;                     B[1][1] = h2 * Mi1;
    B[2][0] = h * Mi0 + h2 * q00 * Mi0; B[2][1] = h2 * q01 * Mi1;
    B[3][0] = h2 * q10 * Mi0;           B[3][1] = h * Mi1 + h2 * q11 * Mi1;
}

// One recurrence step from a staged LDS u-slot.
template <bool WRITE>
__device__ __forceinline__ void do_step(const float2* slot,
                                        const float (&A)[4][4], const float (&B)[4][2],
                                        float& x0, float& x1, float& x2, float& x3,
                                        float2* ostore) {
    const float2 uv = *slot;
    const float n0 = fmaf(A[0][3], x3, fmaf(A[0][2], x2, fmaf(A[0][1], x1,
                      fmaf(A[0][0], x0, B[0][0] * uv.x))));
    const float n1 = fmaf(A[1][3], x3, fmaf(A[1][2], x2, fmaf(A[1][1], x1,
                      fmaf(A[1][0], x0, B[1][1] * uv.y))));
    const float n2 = fmaf(A[2][3], x3, fmaf(A[2][2], x2, fmaf(A[2][1], x1,
                      fmaf(A[2][0], x0, fmaf(B[2][0], uv.x, B[2][1] * uv.y)))));
    const float n3 = fmaf(A[3][3], x3, fmaf(A[3][2], x2, fmaf(A[3][1], x1,
                      fmaf(A[3][0], x0, fmaf(B[3][0], uv.x, B[3][1] * uv.y)))));
    x0 = n0; x1 = n1; x2 = n2; x3 = n3;
    if (WRITE) store_nt_f2(ostore, n0, n1);
}

// Core stepping loop: per-lane async NSLOT-deep ring of u slices staged in LDS.
// ubase = uniform byte address of u[t0][0][0]; voff = lane byte offset b*8
// (< 32KB, fits signed i32). optr -> out[t0][b] (stride BATCH float2).
// slot0 = this lane's slot in ring[0]; consecutive slots are BLK float2 apart.
// l0 = LDS byte offset of slot0; consecutive slots are BLK*8 bytes apart.
template <bool WRITE>
__device__ __forceinline__ void integrate(unsigned long long ubase, unsigned voff,
                                          float2* __restrict__ optr, int nsteps,
                                          const float (&A)[4][4], const float (&B)[4][2],
                                          float& x0, float& x1, float& x2, float& x3,
                                          const float2* slot0, unsigned l0) {
    // Prologue: stage steps 0..NSLOT-2 (NSLOT-1 in flight).
#pragma unroll
    for (int k = 0; k < NSLOT - 1; ++k)
        async_copy_b64_gvs(l0 + (unsigned)k * (BLK * 8u), voff,
                           ubase + (unsigned long long)k * STEP_BYTES);
    int t = 0;
    // Steady state: issue t+NSLOT-1, wait asynccnt<=NSLOT-1 (=> t staged), consume t.
    for (; t < nsteps - (NSLOT - 1); ++t) {
        async_copy_b64_gvs(l0 + (unsigned)((t + NSLOT - 1) & (NSLOT - 1)) * (BLK * 8u),
                           voff, ubase + (unsigned long long)(t + NSLOT - 1) * STEP_BYTES);
        wait_async_steady();
        do_step<WRITE>(slot0 + (size_t)(t & (NSLOT - 1)) * BLK, A, B, x0, x1, x2, x3,
                       optr + (size_t)t * BATCH);
    }
    // Epilogue: drain the pipeline.
    for (int k = NSLOT - 2; k >= 0; --k, ++t) {
        wait_async_le(k);
        do_step<WRITE>(slot0 + (size_t)(t & (NSLOT - 1)) * BLK, A, B, x0, x1, x2, x3,
                       optr + (size_t)t * BATCH);
    }
}

// Pass 1: per (segment, batch) partial with zero start state -> q_s.
// Only segments 0..SEGS-2 are needed (q of the last segment is never consumed).
__global__ void __launch_bounds__(BLK) seg_partial_kernel(const float* __restrict__ u,
                                                          const float* __restrict__ Ct,
                                                          float4* __restrict__ qout) {
    __shared__ float2 ubuf[NSLOT][BLK];
    const int tid = threadIdx.x;
    const int s = blockIdx.x / (BATCH / BLK);
    const int b = (blockIdx.x % (BATCH / BLK)) * BLK + tid;

    float A[4][4], B[4][2];
    compute_AdBd(Ct, A, B);

    float x0 = 0.0f, x1 = 0.0f, x2 = 0.0f, x3 = 0.0f;
    const unsigned long long ubase =
        (unsigned long long)(size_t)(u + (size_t)s * SEG_LEN * BATCH * 2);
    const unsigned voff = (unsigned)b * 8u;
    const unsigned l0 = (unsigned)(size_t)&ubuf[0][tid];

    integrate<false>(ubase, voff, nullptr, SEG_LEN, A, B, x0, x1, x2, x3,
                     &ubuf[0][tid], l0);

    store_nt_f4(&qout[(size_t)s * BATCH + b], x0, x1, x2, x3);
}

// Pass 2: per-batch scan over segments: start_{s+1} = Ad^L start_s + q_s
__global__ void __launch_bounds__(BLK) seg_scan_kernel(const float4* __restrict__ qbuf,
                                                       const float4* __restrict__ x0v,
                                                       const float* __restrict__ Ct,
                                                       float4* __restrict__ starts) {
    const int b = blockIdx.x * BLK + threadIdx.x;

    float A[4][4], B[4][2];
    compute_AdBd(Ct, A, B);

    // M = Ad^SEG_LEN by repeated squaring (SEG_LEN is a power of two)
    float M[4][4];
#pragma unroll
    for (int i = 0; i < 4; ++i)
#pragma unroll
        for (int j = 0; j < 4; ++j) M[i][j] = A[i][j];
    for (int l = SEG_LEN; l > 1; l >>= 1) {
        float Tm[4][4];
#pragma unroll
        for (int i = 0; i < 4; ++i)
#pragma unroll
            for (int j = 0; j < 4; ++j) {
                float acc = 0.0f;
#pragma unroll
                for (int k = 0; k < 4; ++k) acc = fmaf(M[i][k], M[k][j], acc);
                Tm[i][j] = acc;
            }
#pragma unroll
        for (int i = 0; i < 4; ++i)
#pragma unroll
            for (int j = 0; j < 4; ++j) M[i][j] = Tm[i][j];
    }

    float4 x = x0v[b];
    for (int s = 0; s < SEGS; ++s) {
        store_nt_f4(&starts[(size_t)s * BATCH + b], x.x, x.y, x.z, x.w);
        if (s + 1 < SEGS) {
            const float4 q = qbuf[(size_t)s * BATCH + b];
            const float n0 = fmaf(M[0][3], x.w, fmaf(M[0][2], x.z, fmaf(M[0][1], x.y, fmaf(M[0][0], x.x, q.x))));
            const float n1 = fmaf(M[1][3], x.w, fmaf(M[1][2], x.z, fmaf(M[1][1], x.y, fmaf(M[1][0], x.x, q.y))));
            const float n2 = fmaf(M[2][3], x.w, fmaf(M[2][2], x.z, fmaf(M[2][1], x.y, fmaf(M[2][0], x.x, q.z))));
            const float n3 = fmaf(M[3][3], x.w, fmaf(M[3][2], x.z, fmaf(M[3][1], x.y, fmaf(M[3][0], x.x, q.w))));
            x = make_float4(n0, n1, n2, n3);
        }
    }
}

// Pass 3: replay each segment from its true start state, writing outputs
__global__ void __launch_bounds__(BLK) seg_final_kernel(const float* __restrict__ u,
                                                        const float* __restrict__ Ct,
                                                        const float4* __restrict__ starts,
                                                        float2* __restrict__ out) {
    __shared__ float2 ubuf[NSLOT][BLK];
    const int tid = threadIdx.x;
    const int s = blockIdx.x / (BATCH / BLK);
    const int b = (blockIdx.x % (BATCH / BLK)) * BLK + tid;

    float A[4][4], B[4][2];
    compute_AdBd(Ct, A, B);

    const float4 xs = starts[(size_t)s * BATCH + b];
    float x0 = xs.x, x1 = xs.y, x2 = xs.z, x3 = xs.w;
    const size_t t0 = (size_t)s * SEG_LEN;
    const unsigned long long ubase = (unsigned long long)(size_t)(u + t0 * BATCH * 2);
    const unsigned voff = (unsigned)b * 8u;
    float2* optr = out + t0 * BATCH + b;
    const unsigned l0 = (unsigned)(size_t)&ubuf[0][tid];

    integrate<true>(ubase, voff, optr, SEG_LEN, A, B, x0, x1, x2, x3,
                    &ubuf[0][tid], l0);
}

// Fallback: monolithic per-batch scan (used only if d_ws is too small)
__global__ void __launch_bounds__(BLK) full_serial_kernel(const float* __restrict__ u,
                                                          const float4* __restrict__ x0v,
                                                          const float* __restrict__ Ct,
                                                          float2* __restrict__ out) {
    __shared__ float2 ubuf[NSLOT][BLK];
    const int tid = threadIdx.x;
    const int b = blockIdx.x * BLK + tid;

    float A[4][4], B[4][2];
    compute_AdBd(Ct, A, B);

    const float4 xs = x0v[b];
    float x0 = xs.x, x1 = xs.y, x2 = xs.z, x3 = xs.w;
    const unsigned long long ubase = (unsigned long long)(size_t)u;
    const unsigned voff = (unsigned)b * 8u;
    const unsigned l0 = (unsigned)(size_t)&ubuf[0][tid];

    integrate<true>(ubase, voff, out + b, T_STEPS, A, B, x0, x1, x2, x3,
                    &ubuf[0][tid], l0);
}

extern "C" void kernel_launch(void* const* d_in, const int* in_sizes, int n_in,
                              void* d_out, int out_size, void* d_ws, size_t ws_size,
                              hipStream_t stream) {
    const float* u   = (const float*)d_in[0];   // (T, BATCH, 2) fp32
    const float* x0v = (const float*)d_in[1];   // (BATCH, 4)   fp32
    const float* Ct  = (const float*)d_in[2];   // (3,)         fp32
    float2* out = (float2*)d_out;               // (T, BATCH) float2

    const size_t need = (size_t)2 * SEGS * BATCH * sizeof(float4);  // 16 MB
    if (ws_size >= need) {
        float4* qbuf   = (float4*)d_ws;
        float4* starts = qbuf + (size_t)SEGS * BATCH;
        const dim3 blk(BLK);
        // Pass 1 skips the last segment: its q is never consumed by the scan.
        hipLaunchKernelGGL(seg_partial_kernel, dim3((SEGS - 1) * (BATCH / BLK)), blk,
                           0, stream, u, Ct, qbuf);
        hipLaunchKernelGGL(seg_scan_kernel, dim3(BATCH / BLK), blk, 0, stream,
                           qbuf, (const float4*)x0v, Ct, starts);
        hipLaunchKernelGGL(seg_final_kernel, dim3(SEGS * (BATCH / BLK)), blk, 0, stream,
                           u, Ct, starts, out);
    } else {
        hipLaunchKernelGGL(full_serial_kernel, dim3(BATCH / BLK), dim3(BLK), 0, stream,
                           u, (const float4*)x0v, Ct, out);
    }
}